// SSDDecode_31086973289063
// MI455X (gfx1250) — compile-verified
//
#include <hip/hip_runtime.h>
#include <hip/hip_bf16.h>

#define DEV __device__ __forceinline__

typedef unsigned int v4u __attribute__((ext_vector_type(4)));
typedef int          v4i __attribute__((ext_vector_type(4)));
typedef int          v8i __attribute__((ext_vector_type(8)));

constexpr int   B_       = 32;
constexpr int   N_       = 24564;
constexpr int   C_       = 21;
constexpr int   ROW_F    = 33;               // 21 conf + 4 loc + 8 anchor
constexpr int   LOC_OFF  = 21;
constexpr int   TOPK_SEL = 200;
constexpr int   NMS_OUT  = 50;
constexpr int   ROWS_PB  = C_ * NMS_OUT;     // 1050
constexpr int   OUT_K    = 200;
constexpr float CONF_T_  = 0.5f;
constexpr float IOU_T_   = 0.45f;
constexpr float IMG_     = 512.0f;
constexpr int   NTHREADS = 256;
constexpr int   NBUCKET  = 2048;
constexpr int   BND_CAP  = 1024;
constexpr float NEG_BIG  = -1.0e30f;

DEV int bucket_of(float s) {
    // all candidates are in (0.5, 1): exponent fixed at 126, mantissa bits
    // [22:12] give a monotone 2048-bucket key.
    return (int)((__float_as_uint(s) >> 12) & (NBUCKET - 1));
}

// Block argmax over (value desc, index asc) — matches lax.top_k/argmax ties.
DEV void reduce_argmax(float* rv, int* ri, int tid)
{
    for (int s = NTHREADS >> 1; s > 0; s >>= 1) {
        __syncthreads();
        if (tid < s) {
            float ov = rv[tid + s]; int oi = ri[tid + s];
            if (ov > rv[tid] || (ov == rv[tid] && oi < ri[tid])) {
                rv[tid] = ov; ri[tid] = oi;
            }
        }
    }
    __syncthreads();
}

__global__ __launch_bounds__(NTHREADS)
void ssd_nms_class_kernel(const float* __restrict__ pred, float* __restrict__ rows)
{
    const int tid = threadIdx.x;
    const int b = blockIdx.x / C_;
    const int c = blockIdx.x % C_;
    const long long rowBase = (long long)b * N_ * ROW_F;

    // Whole class column staged in LDS via TDM (24564 fp32 = 98 KB; CDNA5 WGP
    // has 320 KB, so total ~122 KB still allows 2 blocks per WGP).
    __shared__ float    scores[N_];
    __shared__ unsigned hist[NBUCKET];
    __shared__ float    rv[NTHREADS];
    __shared__ int      ri[NTHREADS];
    __shared__ float    bndV[BND_CAP];
    __shared__ int      bndI[BND_CAP];
    __shared__ float    candV[TOPK_SEL];
    __shared__ int      candN[TOPK_SEL];
    __shared__ float    bx0[TOPK_SEL], by0[TOPK_SEL], bx1[TOPK_SEL], by1[TOPK_SEL];
    __shared__ float    act[TOPK_SEL];
    __shared__ unsigned cntDef, cntBnd;
    __shared__ int      cutB_s;

    for (int i = tid; i < NBUCKET; i += NTHREADS) hist[i] = 0u;
    if (tid == 0) { cntDef = 0u; cntBnd = 0u; }

    // ---- TDM: DMA conf[b][:, c] (stride 33 floats) compacted into LDS ------
    // 2-D tile: tile_dim0 = 1 element (4B), tile_dim1 = N rows,
    // tensor_dim0_stride = 33 elements.  One wave issues; TENSORcnt tracks.
    if (tid < 32) {
        unsigned long long ga = (unsigned long long)(const void*)(pred + rowBase + c);
        unsigned           la = (unsigned)(unsigned long long)(void*)scores;
        v4u g0 = { 1u,                                        // count=1 (valid user D#)
                   la,                                        // lds_addr [63:32]
                   (unsigned)(ga & 0xFFFFFFFFull),            // global_addr lo
                   (unsigned)((ga >> 32) & 0x1FFFFFFull)      // global_addr hi (57b)
                     | 0x80000000u };                         // type=2 ("image")
        v8i g1 = { (int)0x00020000u,                          // wg_mask=0, data_size=2 (4B)
                   (int)(33u << 16),                          // [63:48] tensor_dim0 lo = 33
                   (int)((unsigned)(N_ & 0xFFFF) << 16),      // [95:80] tensor_dim1 lo = N
                   (int)(1u << 16),                           // [111:96] dim1 hi=0; [127:112] tile_dim0=1
                   (int)N_,                                   // [143:128] tile_dim1=N; tile_dim2=0
                   33,                                        // tensor_dim0_stride lo = 33
                   0,                                         // stride0 hi / dim1_stride lo
                   0 };                                       // dim1_stride hi (unused, 2-D)
        v4i gz = { 0, 0, 0, 0 };
#if defined(__clang_major__) && (__clang_major__ >= 23)
        v8i gz8 = { 0, 0, 0, 0, 0, 0, 0, 0 };
        __builtin_amdgcn_tensor_load_to_lds(g0, g1, gz, gz, gz8, 0);
#else
        __builtin_amdgcn_tensor_load_to_lds(g0, g1, gz, gz, 0);
#endif
        __builtin_amdgcn_s_wait_tensorcnt(0);
    }
    __syncthreads();
    asm volatile("" ::: "memory");

    // ---- pass 1 (LDS): histogram of scores above threshold -----------------
    for (int n = tid; n < N_; n += NTHREADS) {
        float s = scores[n];
        if (s > CONF_T_) atomicAdd(&hist[bucket_of(s)], 1u);
    }
    __syncthreads();

    // ---- find the bucket holding rank-200 ----------------------------------
    if (tid == 0) {
        unsigned cum = 0u; int cb = -1;
        for (int bkt = NBUCKET - 1; bkt >= 0; --bkt) {
            unsigned h = hist[bkt];
            if (cum + h >= (unsigned)TOPK_SEL) { cb = bkt; break; }
            cum += h;
        }
        cutB_s = cb;          // cb == -1  =>  fewer than 200 above threshold
    }
    __syncthreads();
    const int cutB = cutB_s;

    // ---- pass 2 (LDS): compact definite top-k + boundary-bucket items ------
    for (int n = tid; n < N_; n += NTHREADS) {
        float s = scores[n];
        if (s > CONF_T_) {
            int bkt = bucket_of(s);
            if (cutB < 0 || bkt > cutB) {
                unsigned p = atomicAdd(&cntDef, 1u);      // guaranteed < 200
                candV[p] = s; candN[p] = n;
            } else if (bkt == cutB) {
                unsigned p = atomicAdd(&cntBnd, 1u);
                if (p < (unsigned)BND_CAP) { bndV[p] = s; bndI[p] = n; }
            }
        }
    }
    __syncthreads();

    const int defCnt = (int)cntDef;
    const int bn     = ((int)cntBnd < BND_CAP) ? (int)cntBnd : BND_CAP;
    const int needT  = (cutB >= 0) ? (TOPK_SEL - defCnt) : 0;

    // ---- exact refinement of the boundary bucket ---------------------------
    for (int k = 0; k < needT; ++k) {
        float v = NEG_BIG; int gi = 0x7FFFFFFF; int jj = -1;
        for (int j = tid; j < bn; j += NTHREADS) {
            float bv = bndV[j]; int bi = bndI[j];
            if (bv > v || (bv == v && bi < gi)) { v = bv; gi = bi; jj = j; }
        }
        rv[tid] = v; ri[tid] = gi;
        reduce_argmax(rv, ri, tid);
        float bestV = rv[0]; int bestI = ri[0];
        if (jj >= 0 && v == bestV && gi == bestI) bndV[jj] = NEG_BIG;  // winner removes
        if (tid == 0) {
            int p = defCnt + k;
            if (bestI != 0x7FFFFFFF) { candV[p] = bestV; candN[p] = bestI; }
            else                     { candV[p] = -1.0f; candN[p] = 0; }
        }
        __syncthreads();
    }

    const int candCnt = defCnt + needT;
    for (int i = candCnt + tid; i < TOPK_SEL; i += NTHREADS) { candV[i] = -1.0f; candN[i] = 0; }
    __syncthreads();

    // ---- decode only the 200 selected anchors (loc/anchor gather, L2-hot) --
    if (tid < TOPK_SEL) {
        const float* p = pred + rowBase + (long long)candN[tid] * ROW_F + LOC_OFF;
        float l0 = p[0], l1 = p[1], l2 = p[2], l3 = p[3];
        float a0 = p[4], a1 = p[5], a2 = p[6], a3 = p[7];
        float v0 = p[8], v1 = p[9], v2 = p[10], v3 = p[11];
        float cx = l0 * a2 * v0 + a0;
        float cy = l1 * a3 * v1 + a1;
        float w  = expf(l2 * v2) * a2;
        float h  = expf(l3 * v3) * a3;
        bx0[tid] = (cx - 0.5f * w) * IMG_;
        by0[tid] = (cy - 0.5f * h) * IMG_;
        bx1[tid] = (cx + 0.5f * w) * IMG_;
        by1[tid] = (cy + 0.5f * h) * IMG_;
        act[tid] = candV[tid];
    }
    __syncthreads();

    // ---- greedy NMS: 50 block-argmax rounds --------------------------------
    for (int k = 0; k < NMS_OUT; ++k) {
        float v = NEG_BIG; int gi = 0x7FFFFFFF;
        if (tid < TOPK_SEL) { v = act[tid]; gi = tid; }
        rv[tid] = v; ri[tid] = gi;
        reduce_argmax(rv, ri, tid);
        float selV = rv[0]; int sel = ri[0];
        bool valid = selV > 0.0f;
        float sx0 = bx0[sel], sy0 = by0[sel], sx1 = bx1[sel], sy1 = by1[sel];
        if (tid == 0) {
            float* r = rows + ((size_t)b * ROWS_PB + (size_t)c * NMS_OUT + k) * 6;
            if (valid) { r[0] = (float)c; r[1] = candV[sel];
                         r[2] = sx0; r[3] = sy0; r[4] = sx1; r[5] = sy1; }
            else       { r[0] = 0.f; r[1] = 0.f; r[2] = 0.f;
                         r[3] = 0.f; r[4] = 0.f; r[5] = 0.f; }
        }
        if (valid && tid < TOPK_SEL) {
            float xx0 = fmaxf(sx0, bx0[tid]);
            float yy0 = fmaxf(sy0, by0[tid]);
            float xx1 = fminf(sx1, bx1[tid]);
            float yy1 = fminf(sy1, by1[tid]);
            float inter = fmaxf(xx1 - xx0, 0.f) * fmaxf(yy1 - yy0, 0.f);
            float areaS = (sx1 - sx0) * (sy1 - sy0);
            float areaT = (bx1[tid] - bx0[tid]) * (by1[tid] - by0[tid]);
            float iou = inter / (areaS + areaT - inter + 1e-8f);
            if (iou > IOU_T_ || tid == sel) act[tid] = -1.0f;
        }
        __syncthreads();
    }
}

__global__ __launch_bounds__(NTHREADS)
void ssd_final_topk_kernel(const float* __restrict__ rows, float* __restrict__ out)
{
    const int tid = threadIdx.x;
    const int b = blockIdx.x;
    __shared__ float sc[ROWS_PB];
    __shared__ float rv[NTHREADS];
    __shared__ int   ri[NTHREADS];

    for (int i = tid; i < ROWS_PB; i += NTHREADS)
        sc[i] = rows[((size_t)b * ROWS_PB + i) * 6 + 1];
    __syncthreads();

    for (int k = 0; k < OUT_K; ++k) {
        float v = NEG_BIG; int gi = 0x7FFFFFFF;
        for (int i = tid; i < ROWS_PB; i += NTHREADS) {
            float x = sc[i];
            if (x > v || (x == v && i < gi)) { v = x; gi = i; }
        }
        rv[tid] = v; ri[tid] = gi;
        reduce_argmax(rv, ri, tid);
        int sel = ri[0];
        if (tid == 0) sc[sel] = NEG_BIG;
        if (tid < 6)
            out[((size_t)b * OUT_K + k) * 6 + tid] =
                rows[((size_t)b * ROWS_PB + sel) * 6 + tid];
        __syncthreads();
    }
}

extern "C" void kernel_launch(void* const* d_in, const int* in_sizes, int n_in,
                              void* d_out, int out_size, void* d_ws, size_t ws_size,
                              hipStream_t stream)
{
    (void)in_sizes; (void)n_in; (void)out_size; (void)ws_size;
    const float* pred = (const float*)d_in[0];
    float* out  = (float*)d_out;
    float* rows = (float*)d_ws;            // 32*1050*6 floats = 806,400 B

    ssd_nms_class_kernel<<<dim3(B_ * C_), dim3(NTHREADS), 0, stream>>>(pred, rows);
    ssd_final_topk_kernel<<<dim3(B_), dim3(NTHREADS), 0, stream>>>(rows, out);
}